// ExLoss_22780506538270
// MI455X (gfx1250) — compile-verified
//
#include <hip/hip_runtime.h>
#include <hip/hip_bf16.h>
#include <math.h>

// Problem constants (match reference)
#define B_  256
#define F_  2048
#define C_  16384
#define P_  8
#define N_  32
#define T_MUL     1.0f
#define W_BU_C    1.0f
#define W_H_C     10.0f
#define P_MARGIN_C 0.1f
#define N_MARGIN_C 0.1f

typedef __attribute__((ext_vector_type(2))) float v2f;
typedef __attribute__((ext_vector_type(8))) float v8f;

// ---------------- workspace layout (float units) ----------------
// [0, 256)            norms
// [256, 256+65536)    sims (256x256)
// [65792, 66048)      lse per row
// [66048]             bu_acc (float)
// [66049]             hp_sum (float)
// [66050]             hn_sum (float)
// [66051]             any_hp (int)
// [66052]             any_hn (int)
#define WS_NORMS 0
#define WS_SIMS  256
#define WS_LSE   65792
#define WS_ACC   66048

__global__ void init_acc_kernel(float* __restrict__ acc) {
    if (threadIdx.x < 5) {
        acc[threadIdx.x] = 0.0f;  // also zeroes the two int flags (bit pattern 0)
    }
}

__global__ __launch_bounds__(256)
void norms_kernel(const float* __restrict__ inputs, float* __restrict__ norms) {
    __shared__ float red[256];
    const int i = blockIdx.x;
    const float* row = inputs + (size_t)i * F_;
    float s = 0.0f;
    for (int k = threadIdx.x; k < F_; k += 256) {
        float v = row[k];
        s += v * v;
    }
    red[threadIdx.x] = s;
    __syncthreads();
    for (int off = 128; off > 0; off >>= 1) {
        if ((int)threadIdx.x < off) red[threadIdx.x] += red[threadIdx.x + off];
        __syncthreads();
    }
    if (threadIdx.x == 0) norms[i] = sqrtf(red[0]);
}

// One wave per 16x16 output tile. grid = (C/16, B/16), block = 32.
// M[i,j] = sum_k inputs[i,k] * V[j,k]   (outputs = M * T)
__global__ __launch_bounds__(32)
void gemm_wmma_kernel(const float* __restrict__ A,   // [B, F] row-major
                      const float* __restrict__ V,   // [C, F] row-major
                      float* __restrict__ out)       // [B, C] row-major
{
    const int lane = threadIdx.x;
    const int n  = lane & 15;    // column within tile (B/C/D lane stripe)
    const int hi = lane >> 4;    // half-wave select (K pair / M+8)
    const int j0 = blockIdx.x * 16;
    const int i0 = blockIdx.y * 16;

    // A frag: lane holds A[i0+n][k0 + 2*hi + {0,1}]  (32-bit A 16x4 layout)
    // B frag: lane holds V[j0+n][k0 + 2*hi + {0,1}]  (B = V^T, K rows striped)
    const float* arow = A + (size_t)(i0 + n) * F_ + 2 * hi;
    const float* brow = V + (size_t)(j0 + n) * F_ + 2 * hi;

    v8f acc = {};
#pragma unroll 8
    for (int k0 = 0; k0 < F_; k0 += 4) {
        v2f a = *(const v2f*)(arow + k0);
        v2f b = *(const v2f*)(brow + k0);
        acc = __builtin_amdgcn_wmma_f32_16x16x4_f32(
            /*neg_a=*/false, a, /*neg_b=*/false, b,
            /*c_mod=*/(short)0, acc, /*reuse_a=*/false, /*reuse_b=*/false);
    }

    // C/D layout: VGPR v -> row (v + 8*hi), col n
    float* o = out + (size_t)(i0 + 8 * hi) * C_ + (j0 + n);
#pragma unroll
    for (int v = 0; v < 8; ++v) {
        o[(size_t)v * C_] = acc[v] * T_MUL;
    }
}

__global__ __launch_bounds__(256)
void sims_kernel(const float* __restrict__ inputs,
                 const float* __restrict__ norms,
                 float* __restrict__ sims) {
    const int i = blockIdx.x;
    const int j = threadIdx.x;
    const float4* a = (const float4*)(inputs + (size_t)i * F_);
    const float4* b = (const float4*)(inputs + (size_t)j * F_);
    float s = 0.0f;
    for (int k = 0; k < F_ / 4; ++k) {
        float4 x = a[k], y = b[k];
        s += x.x * y.x + x.y * y.y + x.z * y.z + x.w * y.w;
    }
    sims[i * 256 + j] = s / (norms[i] * norms[j]);
}

__global__ __launch_bounds__(256)
void lse_kernel(const float* __restrict__ out, float* __restrict__ lse) {
    __shared__ float red[256];
    const int i = blockIdx.x;
    const float* row = out + (size_t)i * C_;

    float m = -INFINITY;
    for (int j = threadIdx.x; j < C_; j += 256) m = fmaxf(m, row[j]);
    red[threadIdx.x] = m;
    __syncthreads();
    for (int off = 128; off > 0; off >>= 1) {
        if ((int)threadIdx.x < off)
            red[threadIdx.x] = fmaxf(red[threadIdx.x], red[threadIdx.x + off]);
        __syncthreads();
    }
    m = red[0];
    __syncthreads();

    float s = 0.0f;
    for (int j = threadIdx.x; j < C_; j += 256) s += expf(row[j] - m);
    red[threadIdx.x] = s;
    __syncthreads();
    for (int off = 128; off > 0; off >>= 1) {
        if ((int)threadIdx.x < off) red[threadIdx.x] += red[threadIdx.x + off];
        __syncthreads();
    }
    if (threadIdx.x == 0) lse[i] = m + logf(red[0]);
}

// One block of 256 threads; thread i handles batch row i.
__global__ __launch_bounds__(256)
void mine_kernel(const int* __restrict__ targets,
                 const int* __restrict__ ppairs,
                 const int* __restrict__ npairs,
                 const int* __restrict__ indexs,
                 const int* __restrict__ cluster,
                 const float* __restrict__ outputs,
                 const float* __restrict__ norms,
                 const float* __restrict__ sims,
                 const float* __restrict__ lse,
                 float* __restrict__ acc) {
    __shared__ int s_idx[B_];
    const int i = threadIdx.x;
    s_idx[i] = indexs[i];
    __syncthreads();

    const float inv_norm = 1.0f / norms[i];
    const int tgt = targets[i];
    const float* outrow = outputs + (size_t)i * C_;
    const float* simrow = sims + i * 256;

    // ---- positives: 8 in-batch + 1 table ----
    float psims[P_ + 1];
    bool  pmask[P_ + 1];
#pragma unroll
    for (int p = 0; p < P_; ++p) {
        const int pp = ppairs[i * P_ + p];
        int j = 0;
        bool found = false;
        for (int t = 0; t < B_; ++t) {
            if (s_idx[t] == pp) { j = t; found = true; break; }  // first match (argmax)
        }
        psims[p] = simrow[j];
        pmask[p] = found && (pp >= 0);
    }
    {
        const float tp = outrow[tgt] * inv_norm;  // tsims[i, target] (T==1)
        psims[P_] = tp;
        pmask[P_] = (tp != 0.0f);
    }

    // ---- negatives: 32 in-batch + 32 table ----
    float nsims[2 * N_];
    bool  nmask[2 * N_];
#pragma unroll 4
    for (int q = 0; q < N_; ++q) {
        const int np = npairs[i * N_ + q];
        int j = 0;
        bool found = false;
        for (int t = 0; t < B_; ++t) {
            if (s_idx[t] == np) { j = t; found = true; break; }
        }
        nsims[q] = simrow[j];
        nmask[q] = found && (np >= 0);
        const int cid = cluster[np < 0 ? 0 : np];
        const float tn = outrow[cid] * inv_norm;
        nsims[N_ + q] = tn;
        nmask[N_ + q] = (np >= 0) && (tn != 0.0f);
    }

    // ---- thresholds ----
    bool anyn = false;
    float nmax = -INFINITY;
    for (int q = 0; q < 2 * N_; ++q)
        if (nmask[q]) { anyn = true; nmax = fmaxf(nmax, nsims[q]); }
    const float p_thrd = (anyn ? nmax : -3.0f) + P_MARGIN_C;

    bool anyp = false;
    float pmin = INFINITY;
    for (int p = 0; p < P_ + 1; ++p)
        if (pmask[p]) { anyp = true; pmin = fminf(pmin, psims[p]); }
    const float n_thrd = (anyp ? pmin : 3.0f) - N_MARGIN_C;

    // ---- hard pair sums ----
    float hp_sum = 0.0f;
    int hp_any = 0;
    for (int p = 0; p < P_ + 1; ++p) {
        if (pmask[p] && (psims[p] < p_thrd)) {
            hp_any = 1;
            hp_sum += expf(-2.0f * (psims[p] - 0.5f));
        }
    }
    float hn_sum = 0.0f;
    int hn_any = 0;
    for (int q = 0; q < 2 * N_; ++q) {
        if (nmask[q] && (nsims[q] > n_thrd) && (nsims[q] < 0.999999f)) {
            hn_any = 1;
            hn_sum += expf(50.0f * (nsims[q] - 0.5f));
        }
    }

    // ---- global accumulation ----
    const float bu_i = lse[i] - outrow[tgt];  // -logp[i, target]
    atomicAdd(&acc[0], bu_i);
    if (hp_any) {
        atomicAdd(&acc[1], hp_sum);
        atomicOr((int*)&acc[3], 1);
    }
    if (hn_any) {
        atomicAdd(&acc[2], hn_sum);
        atomicOr((int*)&acc[4], 1);
    }
}

__global__ void finalize_kernel(const float* __restrict__ acc, float* __restrict__ loss_out) {
    const float bu = acc[0] / (float)B_;
    const int any_hp = ((const int*)acc)[3];
    const int any_hn = ((const int*)acc)[4];
    const float hp_loss = any_hp ? 0.5f * log1pf(acc[1]) : 0.0f;
    const float hn_loss = any_hn ? (1.0f / 50.0f) * log1pf(acc[2]) : 0.0f;
    loss_out[0] = W_BU_C * bu + W_H_C * (hp_loss + hn_loss);
}

extern "C" void kernel_launch(void* const* d_in, const int* in_sizes, int n_in,
                              void* d_out, int out_size, void* d_ws, size_t ws_size,
                              hipStream_t stream) {
    // setup_inputs() dict order:
    const float* inputs  = (const float*)d_in[0];  // [B, F]
    const int*   targets = (const int*)  d_in[1];  // [B]
    const int*   ppairs  = (const int*)  d_in[2];  // [B, P]
    const int*   npairs  = (const int*)  d_in[3];  // [B, N]
    const int*   indexs  = (const int*)  d_in[4];  // [B]
    const int*   cluster = (const int*)  d_in[5];  // [NS]
    const float* V       = (const float*)d_in[6];  // [C, F]
    // d_in[7] = epoch (unused)

    float* loss_out = (float*)d_out;            // [0] = loss
    float* outputs  = (float*)d_out + 1;        // [1..] = outputs [B, C]

    float* ws     = (float*)d_ws;
    float* norms  = ws + WS_NORMS;
    float* sims   = ws + WS_SIMS;
    float* lse    = ws + WS_LSE;
    float* acc    = ws + WS_ACC;

    init_acc_kernel<<<1, 32, 0, stream>>>(acc);
    norms_kernel<<<B_, 256, 0, stream>>>(inputs, norms);
    {
        dim3 grid(C_ / 16, B_ / 16);
        gemm_wmma_kernel<<<grid, 32, 0, stream>>>(inputs, V, outputs);
    }
    sims_kernel<<<B_, 256, 0, stream>>>(inputs, norms, sims);
    lse_kernel<<<B_, 256, 0, stream>>>(outputs, lse);
    mine_kernel<<<1, B_, 0, stream>>>(targets, ppairs, npairs, indexs, cluster,
                                      outputs, norms, sims, lse, acc);
    finalize_kernel<<<1, 1, 0, stream>>>(acc, loss_out);
}